// HeteAttention_62629213110491
// MI455X (gfx1250) — compile-verified
//
#include <hip/hip_runtime.h>
#include <math.h>

typedef __attribute__((ext_vector_type(2))) float v2f;
typedef __attribute__((ext_vector_type(8))) float v8f;

#define NB   128      // neighbors
#define NF   8        // facets
#define ED   16       // emb dim
#define FD   128      // NF*ED
#define ZROW 132      // padded z row stride (floats): avoids LDS bank conflicts
#define WROW 17       // padded routing-weight row stride (cols 8..15 zeroed for WMMA B)
#define RROW 9        // padded 1/norm row stride
#define EPS  1e-12f

__device__ __forceinline__ float dot4(float4 a, float4 b) {
    return a.x * b.x + a.y * b.y + a.z * b.z + a.w * b.w;
}

__global__ __launch_bounds__(128) void hete_route_kernel(
    const float* __restrict__ feat,   // [B, FD]
    const float* __restrict__ meta,   // [B, NB, FD]
    float* __restrict__ out)          // [B, FD]
{
    __shared__ __align__(16) float z_lds[NB * ZROW];   // raw z, row-padded
    __shared__ __align__(16) float u_lds[FD];          // current normalized u
    __shared__ __align__(16) float x_lds[FD];          // normalized x (residual)
    __shared__ __align__(16) float w_lds[NB * WROW];   // softmax(p)*rnorm, cols 8..15 = 0
    __shared__ float rn_lds[NB * RROW];                // 1/||z[k,f,:]||

    const int tid  = threadIdx.x;
    const int lane = tid & 31;
    const int wv   = tid >> 5;
    const int b    = blockIdx.x;

    // ---- 1. Async DMA: z tile (128x128 f32 = 64KB) global -> LDS (padded rows) ----
    {
        const float* zsrc = meta + (size_t)b * (NB * FD);
        unsigned lds0 = (unsigned)(size_t)(void*)z_lds;  // generic addr low32 = LDS offset
        #pragma unroll
        for (int c = 0; c < 32; ++c) {
            unsigned g   = (unsigned)(c * 128 + tid);    // 16-byte group id
            unsigned k   = g >> 5;                       // neighbor row
            unsigned c4  = g & 31u;                      // 16B chunk within row
            unsigned dst = lds0 + k * (ZROW * 4u) + c4 * 16u;
            unsigned off = g * 16u;
            asm volatile("global_load_async_to_lds_b128 %0, %1, %2"
                         :: "v"(dst), "v"(off), "s"(zsrc)
                         : "memory");
        }
    }

    // ---- 2. x: load + per-facet L2 norm (overlaps with DMA; shuffle reduce) ----
    float xv  = feat[(size_t)b * FD + tid];
    float ssx = xv * xv;
    ssx += __shfl_xor(ssx, 1, 32);
    ssx += __shfl_xor(ssx, 2, 32);
    ssx += __shfl_xor(ssx, 4, 32);
    ssx += __shfl_xor(ssx, 8, 32);
    float xh = xv / fmaxf(sqrtf(ssx), EPS);
    x_lds[tid] = xh;
    u_lds[tid] = xh;
    // zero WMMA B-operand padding columns (persist across iterations)
    #pragma unroll
    for (int f = NF; f < 16; ++f) w_lds[tid * WROW + f] = 0.0f;

    asm volatile("s_wait_asynccnt 0" ::: "memory");
    __syncthreads();

    for (int it = 0; it < 4; ++it) {
        // ===== Phase A (VALU): logits + softmax over facets; thread = neighbor k =====
        {
            const int k = tid;
            const float4* zr = (const float4*)&z_lds[k * ZROW];
            float p[NF], rnv[NF];
            #pragma unroll
            for (int f = 0; f < NF; ++f) {
                float4 z0 = zr[f * 4 + 0], z1 = zr[f * 4 + 1];
                float4 z2 = zr[f * 4 + 2], z3 = zr[f * 4 + 3];
                const float4* uu = (const float4*)&u_lds[f * ED];  // broadcast reads
                float dt = dot4(z0, uu[0]) + dot4(z1, uu[1]) +
                           dot4(z2, uu[2]) + dot4(z3, uu[3]);
                if (it == 0) {  // fuse z-row norm into first logit sweep
                    float sq = dot4(z0, z0) + dot4(z1, z1) +
                               dot4(z2, z2) + dot4(z3, z3);
                    rnv[f] = 1.0f / fmaxf(sqrtf(sq), EPS);
                    rn_lds[k * RROW + f] = rnv[f];
                } else {
                    rnv[f] = rn_lds[k * RROW + f];
                }
                p[f] = dt * rnv[f];   // zhat . uhat
            }
            float mx = p[0];
            #pragma unroll
            for (int f = 1; f < NF; ++f) mx = fmaxf(mx, p[f]);
            float sm = 0.0f;
            #pragma unroll
            for (int f = 0; f < NF; ++f) { p[f] = __expf(p[f] - mx); sm += p[f]; }
            float inv = 1.0f / sm;
            #pragma unroll
            for (int f = 0; f < NF; ++f)
                w_lds[k * WROW + f] = p[f] * inv * rnv[f];  // fold 1/||z|| into weight
        }
        __syncthreads();

        // ===== Phase B (WMMA f32 16x16x4): U[j,c] = sum_k Z^T[j,k] * W[k,c] =====
        // Wave wv owns j-tiles [32*wv, 32*wv+16) and [+16, +32); each tile is one facet.
        {
            const int koff = (lane < 16) ? 0 : 2;  // ISA A/B f32 frag: VGPR0 K={0|2}
            const int m    = lane & 15;            // A row (= j in tile) / B col n
            #pragma unroll
            for (int t = 0; t < 2; ++t) {
                const int jt = wv * 32 + t * 16;   // tile base j; facet = jt>>4
                v8f acc = {};
                #pragma unroll 8
                for (int k0 = 0; k0 < NB; k0 += 4) {
                    v2f a, bb;
                    a.x  = z_lds[(k0 + koff    ) * ZROW + jt + m];  // Z^T[jt+m, k]
                    a.y  = z_lds[(k0 + koff + 1) * ZROW + jt + m];
                    bb.x = w_lds[(k0 + koff    ) * WROW + m];       // W[k, n] (n>=8 -> 0)
                    bb.y = w_lds[(k0 + koff + 1) * WROW + m];
                    acc = __builtin_amdgcn_wmma_f32_16x16x4_f32(
                        false, a, false, bb, (short)0, acc, false, false);
                }
                // Extract column c == facet of this tile, add x residual, renorm.
                const int fj = jt >> 4;
                if (m == fj) {                       // 2 live lanes: fj and fj+16
                    const int rbase = jt + ((lane < 16) ? 0 : 8);
                    float vals[8];
                    float ss = 0.0f;
                    #pragma unroll
                    for (int v = 0; v < 8; ++v) {
                        float uj = acc[v] + x_lds[rbase + v];
                        vals[v] = uj;
                        ss += uj * uj;
                    }
                    if (it < 3) {
                        ss += __shfl_xor(ss, 16, 32);   // partner lane holds rows 8..15
                        float rn = 1.0f / fmaxf(sqrtf(ss), EPS);
                        #pragma unroll
                        for (int v = 0; v < 8; ++v) u_lds[rbase + v] = vals[v] * rn;
                    } else {
                        #pragma unroll
                        for (int v = 0; v < 8; ++v) u_lds[rbase + v] = vals[v];
                    }
                }
            }
        }
        __syncthreads();
    }

    out[(size_t)b * FD + tid] = u_lds[tid];
}

extern "C" void kernel_launch(void* const* d_in, const int* in_sizes, int n_in,
                              void* d_out, int out_size, void* d_ws, size_t ws_size,
                              hipStream_t stream) {
    const float* feat = (const float*)d_in[0];   // [B, 1, 128] f32
    const float* meta = (const float*)d_in[1];   // [B, 1, 128, 128] f32
    float* out = (float*)d_out;                  // [B, 128] f32
    const int B = in_sizes[0] / FD;              // 4096
    hete_route_kernel<<<dim3(B), dim3(128), 0, stream>>>(feat, meta, out);
}